// DeepGraphConvNet_14998025797672
// MI455X (gfx1250) — compile-verified
//
#include <hip/hip_runtime.h>
#include <math.h>

#define N_NODES 100000
#define N_EDGES 1600000
#define DIM     48
#define HIDDEN  128

typedef __attribute__((ext_vector_type(2))) float v2f;
typedef __attribute__((ext_vector_type(8))) float v8f;

// ---------------------------------------------------------------------------
// Zero-fill (aggregation buffer must be zeroed before each scatter pass).
// ---------------------------------------------------------------------------
__global__ void gnn_fill_zero(float* __restrict__ p, size_t n4) {
  size_t i = (size_t)blockIdx.x * blockDim.x + threadIdx.x;
  size_t stride = (size_t)gridDim.x * blockDim.x;
  float4 z = make_float4(0.f, 0.f, 0.f, 0.f);
  for (; i < n4; i += stride) reinterpret_cast<float4*>(p)[i] = z;
}

// ---------------------------------------------------------------------------
// Edge scatter-add: aggr[dst] += feat[src]  (segment_sum equivalent).
// One edge per thread; 12 x float4 gather + 48 hardware f32 atomics
// (global_atomic_add_f32, no CAS loop). Tables are L2-resident
// (19.2 MB << 192 MB L2), so this is L2-atomic-throughput bound.
// ---------------------------------------------------------------------------
__global__ void gnn_scatter_add(const float* __restrict__ feat,
                                const long long* __restrict__ ei,
                                float* __restrict__ aggr) {
  int e = blockIdx.x * blockDim.x + threadIdx.x;
  if (e >= N_EDGES) return;
  long long s = ei[e];
  long long d = ei[(size_t)N_EDGES + e];
  const float4* srow = reinterpret_cast<const float4*>(feat + (size_t)s * DIM);
  float* drow = aggr + (size_t)d * DIM;
#pragma unroll
  for (int c = 0; c < DIM / 4; ++c) {
    float4 v = srow[c];
    unsafeAtomicAdd(drow + 4 * c + 0, v.x);
    unsafeAtomicAdd(drow + 4 * c + 1, v.y);
    unsafeAtomicAdd(drow + 4 * c + 2, v.z);
    unsafeAtomicAdd(drow + 4 * c + 3, v.w);
  }
}

// ---------------------------------------------------------------------------
// WMMA GEMM:  out = A1 @ W1 [+ A2 @ W2] + bias   (optional ReLU)
// A: [N_NODES x 48] row-major, W: [48 x outDim] row-major.
// One wave computes a 16x16 output tile with V_WMMA_F32_16X16X4_F32.
//
// Data flow: weight column-slice (48x16 = 3 KB/matrix) is staged into LDS
// once per block, TRANSPOSED to [col][k] so each lane's B fragment is one
// aligned ds_load_b64. All 12 A fragments (float2 global loads, clause-able
// immediate offsets off one base) and 12 B fragments are preloaded into
// registers, then 12/24 WMMAs issue back-to-back.
//
// Fragment layouts (ISA 7.12.2, wave32):
//   A 16x4 f32 : lane L(<16) holds A[M=L][K=0..1]; lane L+16 holds K=2..3.
//   B 4x16 f32 : v0 lanes0-15 = row K=0, lanes16-31 = row K=2; v1 = K=1/K=3.
//   C/D 16x16  : 8 VGPRs; lane n(<16) col n rows 0..7; lanes16-31 rows 8..15.
// ---------------------------------------------------------------------------
template <bool DUAL, bool RELU>
__global__ void gnn_gemm48_wmma(const float* __restrict__ A1,
                                const float* __restrict__ W1,
                                const float* __restrict__ A2,
                                const float* __restrict__ W2,
                                const float* __restrict__ bias,
                                float* __restrict__ out,
                                int outDim) {
  __shared__ float ldsW1[16 * 48];
  __shared__ float ldsW2[DUAL ? 16 * 48 : 1];

  const int NT = N_NODES / 16;  // 6250 node tiles
  int colBase = blockIdx.y * 16;

  // --- Cooperative staging: weight row k, column colBase + c goes to LDS slot c*48 + k ---
  for (int idx = threadIdx.x; idx < 16 * 48; idx += blockDim.x) {
    int c = idx / 48;
    int k = idx - c * 48;
    ldsW1[idx] = W1[(size_t)k * outDim + colBase + c];
    if (DUAL) ldsW2[idx] = W2[(size_t)k * outDim + colBase + c];
  }
  __syncthreads();

  // Clamp (not return): keeps __syncthreads valid and EXEC all-1s for WMMA.
  // Clamped waves redundantly recompute tile NT-1 and store identical values.
  int nodeTile = blockIdx.x * (blockDim.x >> 5) + (int)(threadIdx.x >> 5);
  if (nodeTile >= NT) nodeTile = NT - 1;

  int lane = threadIdx.x & 31;
  int l15  = lane & 15;
  int kh   = lane >> 4;  // 0: K=0,1 / rows 0..7 ; 1: K=2,3 / rows 8..15
  int nodeBase = nodeTile * 16;
  int col      = colBase + l15;

  const float* a1row = A1 + (size_t)(nodeBase + l15) * DIM;
  const float* a2row = DUAL ? (A2 + (size_t)(nodeBase + l15) * DIM) : nullptr;
  const float* w1row = &ldsW1[l15 * 48];
  const float* w2row = DUAL ? &ldsW2[l15 * 48] : nullptr;

  // --- Preload all fragments (loads clause together, single wait) ---
  v2f fA1[12], fB1[12];
  v2f fA2[DUAL ? 12 : 1], fB2[DUAL ? 12 : 1];
#pragma unroll
  for (int j = 0; j < 12; ++j) {
    int kr = 4 * j + 2 * kh;
    float2 ta = *reinterpret_cast<const float2*>(a1row + kr);
    fA1[j] = {ta.x, ta.y};
    float2 tb = *reinterpret_cast<const float2*>(w1row + kr);  // ds_load_b64
    fB1[j] = {tb.x, tb.y};
    if (DUAL) {
      float2 ua = *reinterpret_cast<const float2*>(a2row + kr);
      fA2[j] = {ua.x, ua.y};
      float2 ub = *reinterpret_cast<const float2*>(w2row + kr);
      fB2[j] = {ub.x, ub.y};
    }
  }

  // --- Back-to-back WMMA MAC chain ---
  v8f c = {};
#pragma unroll
  for (int j = 0; j < 12; ++j) {
    c = __builtin_amdgcn_wmma_f32_16x16x4_f32(false, fA1[j], false, fB1[j],
                                              (short)0, c, false, false);
    if (DUAL)
      c = __builtin_amdgcn_wmma_f32_16x16x4_f32(false, fA2[j], false, fB2[j],
                                                (short)0, c, false, false);
  }

  float bc = bias[col];
#pragma unroll
  for (int r = 0; r < 8; ++r) {
    float v = c[r] + bc;
    if (RELU) v = fmaxf(v, 0.f);
    out[(size_t)(nodeBase + r + 8 * kh) * outDim + col] = v;
  }
}

// ---------------------------------------------------------------------------
// Final 128 -> 1 dot product + sigmoid. One node per thread.
// ---------------------------------------------------------------------------
__global__ void gnn_mlp_out(const float* __restrict__ hidden,
                            const float* __restrict__ w,
                            const float* __restrict__ b,
                            float* __restrict__ out) {
  int i = blockIdx.x * blockDim.x + threadIdx.x;
  if (i >= N_NODES) return;
  const float4* hr = reinterpret_cast<const float4*>(hidden + (size_t)i * HIDDEN);
  const float4* wr = reinterpret_cast<const float4*>(w);
  float acc = 0.f;
#pragma unroll
  for (int j = 0; j < HIDDEN / 4; ++j) {
    float4 hv = hr[j];
    float4 wv = wr[j];
    acc = fmaf(hv.x, wv.x, acc);
    acc = fmaf(hv.y, wv.y, acc);
    acc = fmaf(hv.z, wv.z, acc);
    acc = fmaf(hv.w, wv.w, acc);
  }
  acc += b[0];
  out[i] = 1.0f / (1.0f + __expf(-acc));
}

// ---------------------------------------------------------------------------
extern "C" void kernel_launch(void* const* d_in, const int* in_sizes, int n_in,
                              void* d_out, int out_size, void* d_ws, size_t ws_size,
                              hipStream_t stream) {
  const float*     x     = (const float*)d_in[0];
  const long long* ei    = (const long long*)d_in[1];   // int64 edge_index [2, E]
  const float*     c0_Wr = (const float*)d_in[2];
  const float*     c0_br = (const float*)d_in[3];
  const float*     c0_Wk = (const float*)d_in[4];
  const float*     c1_Wr = (const float*)d_in[5];
  const float*     c1_br = (const float*)d_in[6];
  const float*     c1_Wk = (const float*)d_in[7];
  const float*     c2_Wr = (const float*)d_in[8];
  const float*     c2_br = (const float*)d_in[9];
  const float*     c2_Wk = (const float*)d_in[10];
  const float*     m0_W  = (const float*)d_in[11];
  const float*     m0_b  = (const float*)d_in[12];
  const float*     m1_W  = (const float*)d_in[13];
  const float*     m1_b  = (const float*)d_in[14];
  float* out = (float*)d_out;

  float* ws   = (float*)d_ws;
  float* aggr = ws;                                   // 100000*48
  float* hA   = aggr + (size_t)N_NODES * DIM;         // 100000*48
  float* hB   = hA   + (size_t)N_NODES * DIM;         // 100000*48
  float* hid  = hB   + (size_t)N_NODES * DIM;         // 100000*128

  const size_t aggr4 = (size_t)N_NODES * DIM / 4;
  const int NT = N_NODES / 16;                        // 6250 node tiles
  dim3 gemmBlock(128);                                // 4 waves / block
  dim3 gridConv((NT + 3) / 4, DIM / 16);              // 16-col tiles: 3
  dim3 gridMlp((NT + 3) / 4, HIDDEN / 16);            // 16-col tiles: 8
  int scatterBlocks = (N_EDGES + 255) / 256;

  // ---- GraphConv layer 0: hA = scatter(x)@Wr + br + x@Wk
  gnn_fill_zero<<<512, 256, 0, stream>>>(aggr, aggr4);
  gnn_scatter_add<<<scatterBlocks, 256, 0, stream>>>(x, ei, aggr);
  gnn_gemm48_wmma<true, false><<<gridConv, gemmBlock, 0, stream>>>(
      aggr, c0_Wr, x, c0_Wk, c0_br, hA, DIM);
  // ---- GraphConv layer 1: hB = scatter(hA)@Wr + br + hA@Wk
  gnn_fill_zero<<<512, 256, 0, stream>>>(aggr, aggr4);
  gnn_scatter_add<<<scatterBlocks, 256, 0, stream>>>(hA, ei, aggr);
  gnn_gemm48_wmma<true, false><<<gridConv, gemmBlock, 0, stream>>>(
      aggr, c1_Wr, hA, c1_Wk, c1_br, hB, DIM);
  // ---- GraphConv layer 2: hA = scatter(hB)@Wr + br + hB@Wk
  gnn_fill_zero<<<512, 256, 0, stream>>>(aggr, aggr4);
  gnn_scatter_add<<<scatterBlocks, 256, 0, stream>>>(hB, ei, aggr);
  gnn_gemm48_wmma<true, false><<<gridConv, gemmBlock, 0, stream>>>(
      aggr, c2_Wr, hB, c2_Wk, c2_br, hA, DIM);
  // ---- MLP: hid = relu(hA @ m0_W + m0_b); out = sigmoid(hid @ m1_W + m1_b)
  gnn_gemm48_wmma<false, true><<<gridMlp, gemmBlock, 0, stream>>>(
      hA, m0_W, nullptr, nullptr, m0_b, hid, HIDDEN);
  gnn_mlp_out<<<(N_NODES + 255) / 256, 256, 0, stream>>>(hid, m1_W, m1_b, out);
}